// GraphEncoder_48180943127324
// MI455X (gfx1250) — compile-verified
//
#include <hip/hip_runtime.h>
#include <hip/hip_bf16.h>
#include <math.h>

// ---------------------------------------------------------------------------
// GraphEncoder on MI455X (gfx1250, wave32, WMMA).
// Phase 1: hid = ELU(gather(emb)[98304,512] @ W_eh^T)  -> f32 WMMA 16x16x4
// Phase 2: 48 sequential attention+GRU steps, batch 32 -> skinny f32 WMMA GEMMs
// ---------------------------------------------------------------------------

typedef float v2f __attribute__((ext_vector_type(2)));
typedef float v8f __attribute__((ext_vector_type(8)));

#define SEQ   64
#define HID_  512
#define BSZ_  32
#define SENTS 48
#define NROWS (BSZ_ * SENTS)          // 1536
#define MALL  (NROWS * SEQ)           // 98304 rows of hid

__device__ __forceinline__ v8f wmma_f32(v2f a, v2f b, v8f c) {
    return __builtin_amdgcn_wmma_f32_16x16x4_f32(false, a, false, b, (short)0, c,
                                                 false, false);
}

// ---------------------------------------------------------------------------
// lengths[i] = max(count_nonzero(concept[i,:]), 1)
// ---------------------------------------------------------------------------
__global__ void len_kernel(const int* __restrict__ concept, int* __restrict__ lens) {
    int i = blockIdx.x * 256 + threadIdx.x;
    if (i >= NROWS) return;
    const int* row = concept + (size_t)i * SEQ;
    int c = 0;
    for (int s = 0; s < SEQ; ++s) c += (row[s] != 0) ? 1 : 0;
    lens[i] = c > 0 ? c : 1;
}

// ---------------------------------------------------------------------------
// hid[m, n] = ELU( sum_k emb_table[concept_flat[m], k] * W_eh[n, k] + b_eh[n] )
// Block: 256 threads (8 waves). Tile: 64(M) x 64(N), K-chunks of 32 via LDS.
// Wave w: wm = w&3 (M subtile), wn = w>>2 (32-wide N half); 2 accumulators.
// ---------------------------------------------------------------------------
#define BK 32
__global__ __launch_bounds__(256)
void emb_hid_kernel(const int* __restrict__ concept,
                    const float* __restrict__ emb,   // [32000, 512]
                    const float* __restrict__ W,     // [512, 512]
                    const float* __restrict__ bias,  // [512]
                    float* __restrict__ hid)         // [98304, 512]
{
    __shared__ float As[64][BK + 4];   // stride 36 floats = 144B (16B aligned)
    __shared__ float Bs[64][BK + 4];
    __shared__ int   toks[64];

    const int t  = threadIdx.x;
    const int m0 = blockIdx.x * 64;
    const int n0 = blockIdx.y * 64;

    if (t < 64) toks[t] = concept[m0 + t];
    __syncthreads();

    const int wv   = t >> 5;
    const int lane = t & 31;
    const int wm   = wv & 3;        // 0..3
    const int wn   = wv >> 2;       // 0..1
    const int half = lane >> 4;     // lanes 16-31 carry K+2,K+3
    const int lr   = lane & 15;

    const int lrow = t >> 2;        // 0..63
    const int lcol = (t & 3) * 8;   // 0,8,16,24

    v8f acc0 = {};
    v8f acc1 = {};

    for (int k0 = 0; k0 < HID_; k0 += BK) {
        // Stage gathered A tile and W_eh tile (each 64x32 f32) into LDS.
        {
            const float4* asrc =
                (const float4*)(emb + (size_t)toks[lrow] * HID_ + k0 + lcol);
            *(float4*)&As[lrow][lcol]     = asrc[0];
            *(float4*)&As[lrow][lcol + 4] = asrc[1];
            const float4* wsrc =
                (const float4*)(W + (size_t)(n0 + lrow) * HID_ + k0 + lcol);
            *(float4*)&Bs[lrow][lcol]     = wsrc[0];
            *(float4*)&Bs[lrow][lcol + 4] = wsrc[1];
        }
        __syncthreads();
#pragma unroll
        for (int kk = 0; kk < BK; kk += 4) {
            const int ks = kk + half * 2;
            v2f a;
            a.x = As[wm * 16 + lr][ks];
            a.y = As[wm * 16 + lr][ks + 1];
            v2f b0, b1;
            b0.x = Bs[wn * 32 + lr][ks];
            b0.y = Bs[wn * 32 + lr][ks + 1];
            b1.x = Bs[wn * 32 + 16 + lr][ks];
            b1.y = Bs[wn * 32 + 16 + lr][ks + 1];
            acc0 = wmma_f32(a, b0, acc0);
            acc1 = wmma_f32(a, b1, acc1);
        }
        __syncthreads();
    }

    // Epilogue: C/D layout -> VGPR r holds M = half*8 + r, N = lane&15.
    const int nb = n0 + wn * 32;
#pragma unroll
    for (int r = 0; r < 8; ++r) {
        const int m = m0 + wm * 16 + half * 8 + r;
        int n = nb + lr;
        float v = acc0[r] + bias[n];
        v = v > 0.f ? v : expm1f(v);
        hid[(size_t)m * HID_ + n] = v;
        n = nb + 16 + lr;
        v = acc1[r] + bias[n];
        v = v > 0.f ? v : expm1f(v);
        hid[(size_t)m * HID_ + n] = v;
    }
}

// ---------------------------------------------------------------------------
// out[32, N] = act( A[32, K] @ W[N, K]^T + bias ), act: 0=id, 1=tanh
// 128 threads = 4 waves, one 16x16 WMMA tile per wave. Grid = (2*N/16)/4.
// Per-lane float2 loads == exact 16x16x4 f32 A/B fragment layout.
// ---------------------------------------------------------------------------
__global__ __launch_bounds__(128)
void gemm32_kernel(const float* __restrict__ A, const float* __restrict__ W,
                   const float* __restrict__ bias, float* __restrict__ out,
                   int N, int K, int act)
{
    const int wv   = threadIdx.x >> 5;
    const int lane = threadIdx.x & 31;
    const int tile = blockIdx.x * 4 + wv;
    const int mT   = tile & 1;
    const int nT   = tile >> 1;
    const int half = lane >> 4;
    const int lr   = lane & 15;

    const float* Ar = A + (size_t)(mT * 16 + lr) * K + half * 2;
    const float* Wr = W + (size_t)(nT * 16 + lr) * K + half * 2;

    v8f acc = {};
#pragma unroll 4
    for (int k = 0; k < K; k += 4) {
        const float2 av = *(const float2*)(Ar + k);
        const float2 wv2 = *(const float2*)(Wr + k);
        v2f a; a.x = av.x;  a.y = av.y;
        v2f b; b.x = wv2.x; b.y = wv2.y;
        acc = wmma_f32(a, b, acc);
    }

    const int n = nT * 16 + lr;
    const float bv = bias ? bias[n] : 0.f;
#pragma unroll
    for (int r = 0; r < 8; ++r) {
        const int m = mT * 16 + half * 8 + r;
        float v = acc[r] + bv;
        if (act == 1) v = tanhf(v);
        out[(size_t)m * N + n] = v;
    }
}

// out[32, 512] = tanh( concat(A0,A1)[32,1024] @ W[512,1024]^T )
__global__ __launch_bounds__(128)
void gemm32cat_kernel(const float* __restrict__ A0, const float* __restrict__ A1,
                      const float* __restrict__ W, float* __restrict__ out)
{
    const int wv   = threadIdx.x >> 5;
    const int lane = threadIdx.x & 31;
    const int tile = blockIdx.x * 4 + wv;
    const int mT   = tile & 1;
    const int nT   = tile >> 1;
    const int half = lane >> 4;
    const int lr   = lane & 15;

    const float* Ar0 = A0 + (size_t)(mT * 16 + lr) * HID_ + half * 2;
    const float* Ar1 = A1 + (size_t)(mT * 16 + lr) * HID_ + half * 2;
    const float* Wr  = W + (size_t)(nT * 16 + lr) * (2 * HID_) + half * 2;

    v8f acc = {};
#pragma unroll 4
    for (int k = 0; k < HID_; k += 4) {
        const float2 av = *(const float2*)(Ar0 + k);
        const float2 wv2 = *(const float2*)(Wr + k);
        v2f a; a.x = av.x;  a.y = av.y;
        v2f b; b.x = wv2.x; b.y = wv2.y;
        acc = wmma_f32(a, b, acc);
    }
#pragma unroll 4
    for (int k = 0; k < HID_; k += 4) {
        const float2 av = *(const float2*)(Ar1 + k);
        const float2 wv2 = *(const float2*)(Wr + HID_ + k);
        v2f a; a.x = av.x;  a.y = av.y;
        v2f b; b.x = wv2.x; b.y = wv2.y;
        acc = wmma_f32(a, b, acc);
    }

    const int n = nT * 16 + lr;
#pragma unroll
    for (int r = 0; r < 8; ++r) {
        const int m = mT * 16 + half * 8 + r;
        out[(size_t)m * HID_ + n] = tanhf(acc[r]);
    }
}

// ---------------------------------------------------------------------------
// Masked softmax attention over one sentence step.
// Block b: scores[64] = (q_b . m_b,s), mask pos<len, softmax, c = align @ m.
// ---------------------------------------------------------------------------
__global__ __launch_bounds__(256)
void attn_kernel(const float* __restrict__ hid, const float* __restrict__ q,
                 const int* __restrict__ lens, float* __restrict__ cbuf, int sent)
{
    __shared__ float qsh[HID_];
    __shared__ float wsh[SEQ];
    __shared__ float invs;

    const int b = blockIdx.x;
    const int t = threadIdx.x;
    const float* mbase =
        hid + ((size_t)b * (SENTS * SEQ) + (size_t)sent * SEQ) * HID_;

    qsh[t]       = q[b * HID_ + t];
    qsh[t + 256] = q[b * HID_ + t + 256];
    __syncthreads();

    const int len = lens[b * SENTS + sent];

    if (t < SEQ) {
        float dot = 0.f;
        const float4* mr = (const float4*)(mbase + (size_t)t * HID_);
        const float4* qv = (const float4*)qsh;
#pragma unroll 4
        for (int d = 0; d < HID_ / 4; ++d) {
            const float4 mv = mr[d];
            const float4 qq = qv[d];
            dot += mv.x * qq.x + mv.y * qq.y + mv.z * qq.z + mv.w * qq.w;
        }
        wsh[t] = (t < len) ? dot : -INFINITY;
    }
    __syncthreads();

    if (t == 0) {
        float mx = -INFINITY;
        for (int s = 0; s < SEQ; ++s) mx = fmaxf(mx, wsh[s]);
        float sum = 0.f;
        for (int s = 0; s < SEQ; ++s) {
            const float e = (s < len) ? __expf(wsh[s] - mx) : 0.f;
            wsh[s] = e;
            sum += e;
        }
        invs = 1.f / sum;
    }
    __syncthreads();

    const float inv = invs;
    for (int d = t; d < HID_; d += 256) {
        float acc = 0.f;
#pragma unroll 4
        for (int s = 0; s < SEQ; ++s) acc += wsh[s] * mbase[(size_t)s * HID_ + d];
        cbuf[b * HID_ + d] = acc * inv;
    }
}

// ---------------------------------------------------------------------------
// GRU gate fusion: gi/gh are [32, 1536] = [r | z | n] chunks.
// ---------------------------------------------------------------------------
__global__ __launch_bounds__(256)
void gru_gate_kernel(const float* __restrict__ gi, const float* __restrict__ gh,
                     const float* __restrict__ h, float* __restrict__ hout)
{
    const int idx = blockIdx.x * 256 + threadIdx.x;   // 0..16383
    const int b = idx >> 9;
    const int j = idx & (HID_ - 1);
    const float* gib = gi + (size_t)b * (3 * HID_);
    const float* ghb = gh + (size_t)b * (3 * HID_);
    const float r = 1.f / (1.f + __expf(-(gib[j] + ghb[j])));
    const float z = 1.f / (1.f + __expf(-(gib[HID_ + j] + ghb[HID_ + j])));
    const float n = tanhf(gib[2 * HID_ + j] + r * ghb[2 * HID_ + j]);
    hout[idx] = (1.f - z) * n + z * h[idx];
}

__global__ void h_init_kernel(const float* __restrict__ h0, float* __restrict__ h) {
    const int idx = blockIdx.x * 256 + threadIdx.x;   // 16384
    h[idx] = h0[idx & (HID_ - 1)];
}

// ---------------------------------------------------------------------------
extern "C" void kernel_launch(void* const* d_in, const int* in_sizes, int n_in,
                              void* d_out, int out_size, void* d_ws, size_t ws_size,
                              hipStream_t stream) {
    const int*   concept = (const int*)d_in[0];
    // d_in[1] = ex_segs (scalar 48, compile-time constant here)
    const float* emb_tab = (const float*)d_in[2];
    const float* W_eh    = (const float*)d_in[3];
    const float* b_eh    = (const float*)d_in[4];
    const float* h0      = (const float*)d_in[5];
    const float* W_in    = (const float*)d_in[6];
    const float* W_out   = (const float*)d_in[7];
    const float* W_ih    = (const float*)d_in[8];
    const float* W_hh    = (const float*)d_in[9];
    const float* b_ih    = (const float*)d_in[10];
    const float* b_hh    = (const float*)d_in[11];

    // Workspace carve-up (all chunks 256B-multiples).
    char* p = (char*)d_ws;
    float* hid  = (float*)p; p += (size_t)MALL * HID_ * sizeof(float);   // 192 MiB
    int*   lens = (int*)p;   p += 2048 * sizeof(int);
    float* hA   = (float*)p; p += BSZ_ * HID_ * sizeof(float);
    float* hB   = (float*)p; p += BSZ_ * HID_ * sizeof(float);
    float* qb   = (float*)p; p += BSZ_ * HID_ * sizeof(float);
    float* cb   = (float*)p; p += BSZ_ * HID_ * sizeof(float);
    float* xb   = (float*)p; p += BSZ_ * HID_ * sizeof(float);
    float* gi   = (float*)p; p += BSZ_ * 3 * HID_ * sizeof(float);
    float* gh   = (float*)p; p += BSZ_ * 3 * HID_ * sizeof(float);

    len_kernel<<<(NROWS + 255) / 256, 256, 0, stream>>>(concept, lens);
    emb_hid_kernel<<<dim3(MALL / 64, HID_ / 64), 256, 0, stream>>>(
        concept, emb_tab, W_eh, b_eh, hid);
    h_init_kernel<<<(BSZ_ * HID_) / 256, 256, 0, stream>>>(h0, hA);

    float* hs[2] = {hA, hB};
    for (int step = 0; step < SENTS; ++step) {
        const float* h = hs[step & 1];
        float* hnext = (step == SENTS - 1) ? (float*)d_out : hs[(step + 1) & 1];

        // q = h @ W_in^T                  [32,512] x [512,512]^T
        gemm32_kernel<<<(2 * HID_ / 16) / 4, 128, 0, stream>>>(
            h, W_in, nullptr, qb, HID_, HID_, 0);
        // masked softmax attention -> context c
        attn_kernel<<<BSZ_, 256, 0, stream>>>(hid, qb, lens, cb, step);
        // x = tanh([c,h] @ W_out^T)       K = 1024
        gemm32cat_kernel<<<(2 * HID_ / 16) / 4, 128, 0, stream>>>(cb, h, W_out, xb);
        // gi = x @ W_ih^T + b_ih          [32,1536]
        gemm32_kernel<<<(2 * (3 * HID_) / 16) / 4, 128, 0, stream>>>(
            xb, W_ih, b_ih, gi, 3 * HID_, HID_, 0);
        // gh = h @ W_hh^T + b_hh
        gemm32_kernel<<<(2 * (3 * HID_) / 16) / 4, 128, 0, stream>>>(
            h, W_hh, b_hh, gh, 3 * HID_, HID_, 0);
        // fused GRU gates
        gru_gate_kernel<<<(BSZ_ * HID_) / 256, 256, 0, stream>>>(gi, gh, h, hnext);
    }
}